// EdgeConv_2439541424179
// MI455X (gfx1250) — compile-verified
//
#include <hip/hip_runtime.h>
#include <hip/hip_bf16.h>

typedef float v2f __attribute__((ext_vector_type(2)));
typedef float v8f __attribute__((ext_vector_type(8)));

#define BATCH   8
#define NPTS    1024
#define KNN     20
#define FS      256          // feature row stride (max Cpad)
#define FUSE    512
#define EMB     1024
#define EPSV    1e-5f
#define SLOPE   0.2f
#define ROWS_E  (BATCH*NPTS*KNN)   // 163840 edge rows
#define MT_E    (ROWS_E/16)        // 10240 M-tiles
#define S_EDGE  320                // splits -> 32 tiles each
#define MT_F    ((BATCH*NPTS)/16)  // 512 M-tiles for final GEMM
#define S_FIN   128                // splits -> 4 tiles each
#define NEG_INF (-3.402823e38f)

static __device__ __forceinline__ v8f wmma_f32(v2f a, v2f b, v8f c) {
  // D = A(16x4 f32) * B(4x16 f32) + C(16x16 f32)
  return __builtin_amdgcn_wmma_f32_16x16x4_f32(false, a, false, b, (short)0, c,
                                               false, false);
}

// ---------- feature prep ----------
__global__ void pad_x_kernel(const float* __restrict__ x, float* __restrict__ feat) {
  int t = blockIdx.x * blockDim.x + threadIdx.x;
  if (t >= BATCH * NPTS) return;
  float* f = feat + (size_t)t * FS;
  const float* xr = x + (size_t)t * 3;
  f[0] = xr[0]; f[1] = xr[1]; f[2] = xr[2]; f[3] = 0.f;
}

__global__ void rownorm_kernel(const float* __restrict__ feat, float* __restrict__ xx,
                               int Cpad) {
  int t = blockIdx.x * blockDim.x + threadIdx.x;
  if (t >= BATCH * NPTS) return;
  const float* f = feat + (size_t)t * FS;
  float s = 0.f;
  for (int c = 0; c < Cpad; ++c) { float v = f[c]; s += v * v; }
  xx[t] = s;
}

// ---------- Gram matrix via fp32 WMMA, 2x2 register blocking ----------
// neg_dist = 2*x.xT - xx_i - xx_j, one wave computes a 32x32 output block.
__global__ void gram_wmma_kernel(const float* __restrict__ feat,
                                 const float* __restrict__ xx,
                                 float* __restrict__ dist, int Cpad) {
  int lane = threadIdx.x;
  int half = lane >> 4;
  int l    = lane & 15;
  int j0 = blockIdx.x * 32;
  int i0 = blockIdx.y * 32;
  int b  = blockIdx.z;
  const float* fb = feat + (size_t)b * NPTS * FS;
  v8f c00 = {}, c01 = {}, c10 = {}, c11 = {};
  for (int c0 = 0; c0 < Cpad; c0 += 4) {
    int kb = c0 + half * 2;
    const float* a0p = fb + (size_t)(i0 + l) * FS + kb;
    const float* a1p = fb + (size_t)(i0 + 16 + l) * FS + kb;
    const float* b0p = fb + (size_t)(j0 + l) * FS + kb;
    const float* b1p = fb + (size_t)(j0 + 16 + l) * FS + kb;
    v2f a0; a0.x = a0p[0]; a0.y = a0p[1];
    v2f a1; a1.x = a1p[0]; a1.y = a1p[1];
    v2f b0; b0.x = b0p[0]; b0.y = b0p[1];
    v2f b1; b1.x = b1p[0]; b1.y = b1p[1];
    c00 = wmma_f32(a0, b0, c00);
    c01 = wmma_f32(a0, b1, c01);
    c10 = wmma_f32(a1, b0, c10);
    c11 = wmma_f32(a1, b1, c11);
  }
  const float* xxb = xx + b * NPTS;
  float xj0 = xxb[j0 + l];
  float xj1 = xxb[j0 + 16 + l];
  float* db = dist + (size_t)b * NPTS * NPTS;
#pragma unroll
  for (int v = 0; v < 8; ++v) {
    int M = v + half * 8;
    float xi0 = xxb[i0 + M];
    float xi1 = xxb[i0 + 16 + M];
    db[(size_t)(i0 + M) * NPTS + (j0 + l)]           = 2.f * c00[v] - xi0 - xj0;
    db[(size_t)(i0 + M) * NPTS + (j0 + 16 + l)]      = 2.f * c01[v] - xi0 - xj1;
    db[(size_t)(i0 + 16 + M) * NPTS + (j0 + l)]      = 2.f * c10[v] - xi1 - xj0;
    db[(size_t)(i0 + 16 + M) * NPTS + (j0 + 16 + l)] = 2.f * c11[v] - xi1 - xj1;
  }
}

// ---------- top-k (largest neg_dist) per row, one wave32 per row ----------
__global__ void topk_kernel(const float* __restrict__ dist, int* __restrict__ idxb) {
  int lane = threadIdx.x & 31;
  int wave = threadIdx.x >> 5;
  int p = blockIdx.x * 4 + wave;              // row in [0, B*N)
  const float* row = dist + (size_t)p * NPTS;
  float v[KNN]; int id[KNN];
#pragma unroll
  for (int t = 0; t < KNN; ++t) { v[t] = NEG_INF; id[t] = 0; }
  for (int it = 0; it < NPTS / 32; ++it) {
    int j = it * 32 + lane;
    float cv = row[j];
    int   ci = j;
    if (cv > v[KNN - 1]) {
#pragma unroll
      for (int t = 0; t < KNN; ++t) {
        if (cv > v[t]) {
          float tv = v[t]; int ti = id[t];
          v[t] = cv; id[t] = ci; cv = tv; ci = ti;
        }
      }
    }
  }
  int* orow = idxb + (size_t)p * KNN;
  for (int r = 0; r < KNN; ++r) {
    float m = v[0];
#pragma unroll
    for (int off = 16; off; off >>= 1) m = fmaxf(m, __shfl_xor(m, off, 32));
    unsigned long long bal = __ballot(v[0] == m);
    int win = __ffsll(bal) - 1;
    int gidx = __shfl(id[0], win, 32);
    if (lane == 0) orow[r] = gidx;
    if (lane == win) {
#pragma unroll
      for (int t = 0; t < KNN - 1; ++t) { v[t] = v[t + 1]; id[t] = id[t + 1]; }
      v[KNN - 1] = NEG_INF;
    }
  }
}

// ---------- edge-feature / weight element loaders ----------
__device__ __forceinline__ float eload(const float* __restrict__ feat, int rn, int rj,
                                       int cp, int Cpad) {
  if (cp < Cpad) return feat[rj + cp] - feat[rn + cp];  // neighbor - center
  return feat[rn + (cp - Cpad)];                        // center
}
__device__ __forceinline__ float wload(const float* __restrict__ W, int o, int cp,
                                       int C, int Cpad) {
  if (cp < Cpad) return (cp < C) ? W[o * 2 * C + cp] : 0.f;
  int c2 = cp - Cpad;
  return (c2 < C) ? W[o * 2 * C + C + c2] : 0.f;
}

// ---------- pass 1: edge GEMM + per-channel sum / sum^2 partials ----------
// One wave covers 16 rows x 64 output channels: gathered A reused across 4 WMMAs.
__global__ void edge_stats_wmma(const float* __restrict__ feat,
                                const int* __restrict__ idxb,
                                const float* __restrict__ W,
                                float* __restrict__ partial,
                                int C, int Cpad, int out) {
  int lane = threadIdx.x;
  int half = lane >> 4;
  int l    = lane & 15;
  int nb = blockIdx.x;   // N-block (out/64)
  int s  = blockIdx.y;   // split
  int o0 = nb * 64;
  int K2 = 2 * Cpad;
  float sum[4] = {0.f, 0.f, 0.f, 0.f};
  float sq[4]  = {0.f, 0.f, 0.f, 0.f};
  for (int mt = s; mt < MT_E; mt += S_EDGE) {
    int r  = mt * 16 + l;
    int kk = r % KNN;
    int n  = (r / KNN) & (NPTS - 1);
    int b  = r / (KNN * NPTS);
    int base = ((b << 10) + n) * FS;
    int j = idxb[(size_t)((b << 10) + n) * KNN + kk];
    int basej = ((b << 10) + j) * FS;
    v8f acc[4] = {{}, {}, {}, {}};
    for (int cc = 0; cc < K2; cc += 4) {
      int kb = cc + half * 2;
      v2f a; a.x = eload(feat, base, basej, kb, Cpad);
             a.y = eload(feat, base, basej, kb + 1, Cpad);
#pragma unroll
      for (int t = 0; t < 4; ++t) {
        int o = o0 + t * 16 + l;
        v2f w; w.x = wload(W, o, kb, C, Cpad);
               w.y = wload(W, o, kb + 1, C, Cpad);
        acc[t] = wmma_f32(a, w, acc[t]);
      }
    }
#pragma unroll
    for (int t = 0; t < 4; ++t) {
#pragma unroll
      for (int v = 0; v < 8; ++v) { float y = acc[t][v]; sum[t] += y; sq[t] += y * y; }
    }
  }
#pragma unroll
  for (int t = 0; t < 4; ++t) {
    float su = sum[t] + __shfl_xor(sum[t], 16, 32);
    float sv = sq[t]  + __shfl_xor(sq[t], 16, 32);
    if (lane < 16) {
      int ct = nb * 4 + t;   // channel tile index
      float* slot = partial + (size_t)(ct * S_EDGE + s) * 32;
      slot[l] = su; slot[16 + l] = sv;
    }
  }
}

// ---------- fold partials into BN scale/shift (deterministic fixed-order) ----------
__global__ void bn_finalize_kernel(const float* __restrict__ partial,
                                   const float* __restrict__ g,
                                   const float* __restrict__ bb,
                                   float* __restrict__ scsh, int out, int S,
                                   float invcnt) {
  int o = blockIdx.x * blockDim.x + threadIdx.x;
  if (o >= out) return;
  int nt = o >> 4, l = o & 15;
  float sum = 0.f, sq = 0.f;
  for (int s = 0; s < S; ++s) {
    const float* slot = partial + (size_t)(nt * S + s) * 32;
    sum += slot[l]; sq += slot[16 + l];
  }
  float mean = sum * invcnt;
  float var  = sq * invcnt - mean * mean;
  float sc = g[o] * rsqrtf(var + EPSV);
  scsh[o]       = sc;
  scsh[EMB + o] = bb[o] - mean * sc;
}

// ---------- pass 2: recompute edge GEMM, affine + LReLU + max over k ----------
// One wave: one point (20 neighbor rows, padded to 32) x 64 output channels.
__global__ void edge_apply_wmma(const float* __restrict__ feat,
                                const int* __restrict__ idxb,
                                const float* __restrict__ W,
                                const float* __restrict__ scsh,
                                float* __restrict__ featO,
                                float* __restrict__ fused,
                                int C, int Cpad, int out, int foff) {
  int lane = threadIdx.x;
  int half = lane >> 4;
  int l    = lane & 15;
  int p  = blockIdx.x;          // b*N + n
  int nb = blockIdx.y;          // out/64
  int o0 = nb * 64;
  int b  = p >> 10;
  int base = p * FS;
  int j0 = idxb[(size_t)p * KNN + l];            // tile0 rows: kk = 0..15
  int kk1 = 16 + l;
  bool val1 = (kk1 < KNN);                       // tile1 rows: kk = 16..19 valid
  int j1 = val1 ? idxb[(size_t)p * KNN + kk1] : 0;
  int basej0 = ((b << 10) + j0) * FS;
  int basej1 = ((b << 10) + j1) * FS;
  int K2 = 2 * Cpad;
  v8f lo[4] = {{}, {}, {}, {}};
  v8f hi[4] = {{}, {}, {}, {}};
  for (int cc = 0; cc < K2; cc += 4) {
    int kb = cc + half * 2;
    v2f al; al.x = eload(feat, base, basej0, kb, Cpad);
            al.y = eload(feat, base, basej0, kb + 1, Cpad);
    v2f ah; ah.x = val1 ? eload(feat, base, basej1, kb, Cpad)     : 0.f;
            ah.y = val1 ? eload(feat, base, basej1, kb + 1, Cpad) : 0.f;
#pragma unroll
    for (int t = 0; t < 4; ++t) {
      int o = o0 + t * 16 + l;
      v2f w; w.x = wload(W, o, kb, C, Cpad);
             w.y = wload(W, o, kb + 1, C, Cpad);
      lo[t] = wmma_f32(al, w, lo[t]);
      hi[t] = wmma_f32(ah, w, hi[t]);
    }
  }
#pragma unroll
  for (int t = 0; t < 4; ++t) {
    int o = o0 + t * 16 + l;
    float sc = scsh[o], sh = scsh[EMB + o];
    float m = NEG_INF;
#pragma unroll
    for (int v = 0; v < 8; ++v) {
      float y = lo[t][v] * sc + sh;
      y = (y > 0.f) ? y : SLOPE * y;
      m = fmaxf(m, y);
    }
    if (half == 0) {   // hi valid rows: kk = 16 + v (v < 4) only in lower half
#pragma unroll
      for (int v = 0; v < 4; ++v) {
        float y = hi[t][v] * sc + sh;
        y = (y > 0.f) ? y : SLOPE * y;
        m = fmaxf(m, y);
      }
    }
    m = fmaxf(m, __shfl_xor(m, 16, 32));
    if (lane < 16) {
      featO[(size_t)p * FS + o] = m;
      fused[(size_t)p * FUSE + foff + o] = m;
    }
  }
}

// ---------- final 512 -> 1024 GEMM: stats pass (A reused across 4 N-tiles) ----------
__global__ void final_stats_wmma(const float* __restrict__ fused,
                                 const float* __restrict__ Wf,
                                 float* __restrict__ partial) {
  int lane = threadIdx.x;
  int half = lane >> 4;
  int l    = lane & 15;
  int nb = blockIdx.x;   // EMB/64 = 16 N-blocks
  int s  = blockIdx.y;   // S_FIN splits
  int o0 = nb * 64;
  float sum[4] = {0.f, 0.f, 0.f, 0.f};
  float sq[4]  = {0.f, 0.f, 0.f, 0.f};
  for (int mt = s; mt < MT_F; mt += S_FIN) {
    int r = mt * 16 + l;
    v8f acc[4] = {{}, {}, {}, {}};
    for (int c0 = 0; c0 < FUSE; c0 += 4) {
      int kb = c0 + half * 2;
      v2f a; a.x = fused[(size_t)r * FUSE + kb];
             a.y = fused[(size_t)r * FUSE + kb + 1];
#pragma unroll
      for (int t = 0; t < 4; ++t) {
        int o = o0 + t * 16 + l;
        v2f w; w.x = Wf[(size_t)o * FUSE + kb];
               w.y = Wf[(size_t)o * FUSE + kb + 1];
        acc[t] = wmma_f32(a, w, acc[t]);
      }
    }
#pragma unroll
    for (int t = 0; t < 4; ++t) {
#pragma unroll
      for (int v = 0; v < 8; ++v) { float y = acc[t][v]; sum[t] += y; sq[t] += y * y; }
    }
  }
#pragma unroll
  for (int t = 0; t < 4; ++t) {
    float su = sum[t] + __shfl_xor(sum[t], 16, 32);
    float sv = sq[t]  + __shfl_xor(sq[t], 16, 32);
    if (lane < 16) {
      int ct = nb * 4 + t;
      float* slot = partial + (size_t)(ct * S_FIN + s) * 32;
      slot[l] = su; slot[16 + l] = sv;
    }
  }
}

// ---------- final GEMM apply: 2x2 blocking, affine + LReLU -> d_out ----------
__global__ void final_apply_wmma(const float* __restrict__ fused,
                                 const float* __restrict__ Wf,
                                 const float* __restrict__ scsh,
                                 float* __restrict__ outp) {
  int lane = threadIdx.x;
  int half = lane >> 4;
  int l    = lane & 15;
  int r0 = blockIdx.x * 32;   // 256 M-blocks
  int o0 = blockIdx.y * 32;   // 32 N-blocks
  v8f c00 = {}, c01 = {}, c10 = {}, c11 = {};
  for (int c0 = 0; c0 < FUSE; c0 += 4) {
    int kb = c0 + half * 2;
    v2f a0; a0.x = fused[(size_t)(r0 + l) * FUSE + kb];
            a0.y = fused[(size_t)(r0 + l) * FUSE + kb + 1];
    v2f a1; a1.x = fused[(size_t)(r0 + 16 + l) * FUSE + kb];
            a1.y = fused[(size_t)(r0 + 16 + l) * FUSE + kb + 1];
    v2f w0; w0.x = Wf[(size_t)(o0 + l) * FUSE + kb];
            w0.y = Wf[(size_t)(o0 + l) * FUSE + kb + 1];
    v2f w1; w1.x = Wf[(size_t)(o0 + 16 + l) * FUSE + kb];
            w1.y = Wf[(size_t)(o0 + 16 + l) * FUSE + kb + 1];
    c00 = wmma_f32(a0, w0, c00);
    c01 = wmma_f32(a0, w1, c01);
    c10 = wmma_f32(a1, w0, c10);
    c11 = wmma_f32(a1, w1, c11);
  }
  float sc0 = scsh[o0 + l],      sh0 = scsh[EMB + o0 + l];
  float sc1 = scsh[o0 + 16 + l], sh1 = scsh[EMB + o0 + 16 + l];
#pragma unroll
  for (int v = 0; v < 8; ++v) {
    int M = v + half * 8;
    int ra = r0 + M, rb = r0 + 16 + M;
    float y;
    y = c00[v] * sc0 + sh0; y = (y > 0.f) ? y : SLOPE * y;
    outp[(size_t)ra * EMB + o0 + l] = y;
    y = c01[v] * sc1 + sh1; y = (y > 0.f) ? y : SLOPE * y;
    outp[(size_t)ra * EMB + o0 + 16 + l] = y;
    y = c10[v] * sc0 + sh0; y = (y > 0.f) ? y : SLOPE * y;
    outp[(size_t)rb * EMB + o0 + l] = y;
    y = c11[v] * sc1 + sh1; y = (y > 0.f) ? y : SLOPE * y;
    outp[(size_t)rb * EMB + o0 + 16 + l] = y;
  }
}

extern "C" void kernel_launch(void* const* d_in, const int* in_sizes, int n_in,
                              void* d_out, int out_size, void* d_ws, size_t ws_size,
                              hipStream_t stream) {
  const float* x  = (const float*)d_in[0];
  const float* W[4]  = {(const float*)d_in[1], (const float*)d_in[4],
                        (const float*)d_in[7], (const float*)d_in[10]};
  const float* g[4]  = {(const float*)d_in[2], (const float*)d_in[5],
                        (const float*)d_in[8], (const float*)d_in[11]};
  const float* bb[4] = {(const float*)d_in[3], (const float*)d_in[6],
                        (const float*)d_in[9], (const float*)d_in[12]};
  const float* Wf = (const float*)d_in[13];
  const float* gf = (const float*)d_in[14];
  const float* bf = (const float*)d_in[15];
  float* outp = (float*)d_out;

  // workspace carve (floats), ~69 MB total
  float* ws = (float*)d_ws;
  float* featA   = ws;                               // 8192*256
  float* featB   = featA + (size_t)BATCH*NPTS*FS;    // 8192*256
  float* fused   = featB + (size_t)BATCH*NPTS*FS;    // 8192*512
  float* dist    = fused + (size_t)BATCH*NPTS*FUSE;  // 8*1024*1024
  float* xx      = dist  + (size_t)BATCH*NPTS*NPTS;  // 8192
  float* partial = xx + BATCH*NPTS;                  // 64*128*32 (>= edge need)
  float* scsh    = partial + (size_t)64*S_FIN*32;    // 2*1024
  int*   idxb    = (int*)(scsh + 2*EMB);             // 8192*20

  pad_x_kernel<<<(BATCH*NPTS + 127)/128, 128, 0, stream>>>(x, featA);

  const int Cin[4]  = {3, 64, 64, 128};
  const int Cp[4]   = {4, 64, 64, 128};
  const int Out[4]  = {64, 64, 128, 256};
  const int Foff[4] = {0, 64, 128, 256};
  float* fc = featA;
  float* fo = featB;
  for (int blk = 0; blk < 4; ++blk) {
    rownorm_kernel<<<(BATCH*NPTS + 127)/128, 128, 0, stream>>>(fc, xx, Cp[blk]);
    dim3 gg(NPTS/32, NPTS/32, BATCH);
    gram_wmma_kernel<<<gg, 32, 0, stream>>>(fc, xx, dist, Cp[blk]);
    topk_kernel<<<(BATCH*NPTS)/4, 128, 0, stream>>>(dist, idxb);
    dim3 gs(Out[blk]/64, S_EDGE);
    edge_stats_wmma<<<gs, 32, 0, stream>>>(fc, idxb, W[blk], partial,
                                           Cin[blk], Cp[blk], Out[blk]);
    bn_finalize_kernel<<<(Out[blk] + 63)/64, 64, 0, stream>>>(
        partial, g[blk], bb[blk], scsh, Out[blk], S_EDGE, 1.f / (float)ROWS_E);
    dim3 ga(BATCH*NPTS, Out[blk]/64);
    edge_apply_wmma<<<ga, 32, 0, stream>>>(fc, idxb, W[blk], scsh, fo, fused,
                                           Cin[blk], Cp[blk], Out[blk], Foff[blk]);
    float* t = fc; fc = fo; fo = t;
  }

  dim3 g1(EMB/64, S_FIN);
  final_stats_wmma<<<g1, 32, 0, stream>>>(fused, Wf, partial);
  bn_finalize_kernel<<<EMB/64, 64, 0, stream>>>(partial, gf, bf, scsh, EMB, S_FIN,
                                                1.f / (float)(BATCH*NPTS));
  dim3 g2(MT_F/2, EMB/32);
  final_apply_wmma<<<g2, 32, 0, stream>>>(fused, Wf, scsh, outp);
}